// DADMMLRSame_17368847745614
// MI455X (gfx1250) — compile-verified
//
#include <hip/hip_runtime.h>

// ---------------- problem constants ----------------
#define PN   64           // nodes
#define BSZ  256          // batch per node
#define NF   784          // feature dim
#define KIT  5            // unfolded iterations
#define NODE_ELEMS (BSZ * NF)        // 200704
#define A_ELEMS    (PN * NODE_ELEMS) // 12845056
#define W_ELEMS    (PN * BSZ)        // 16384

typedef float v2f __attribute__((ext_vector_type(2)));
typedef float v8f __attribute__((ext_vector_type(8)));

// ---------------- init: a = a0, w = w0, mu = 0, lam = 0 ----------------
__global__ __launch_bounds__(256)
void dadmm_init(const float* __restrict__ a0, const float* __restrict__ w0,
                float* __restrict__ a, float* __restrict__ w,
                float* __restrict__ mu, float* __restrict__ lam) {
    const int i = blockIdx.x * 256 + threadIdx.x;   // float4 index < A_ELEMS/4
    ((float4*)a)[i]  = ((const float4*)a0)[i];
    ((float4*)mu)[i] = make_float4(0.f, 0.f, 0.f, 0.f);
    if (i < W_ELEMS / 4) {
        ((float4*)w)[i]   = ((const float4*)w0)[i];
        ((float4*)lam)[i] = make_float4(0.f, 0.f, 0.f, 0.f);
    }
}

// ---------------- Gauss-Seidel color sweep ----------------
// grid = (16 batch-groups, 32 nodes-of-color), block = 128 (4 waves).
// Each block: 16 rows (batches) of one node.
//   Phase 1: xta for 16 rows via V_WMMA_F32_16X16X4_F32 (4 waves split K=784).
//            Operand loads use immediate-offset global_load_b64 (pointer is
//            bumped once per 7 chunks) so the inner loop is loads+wmma only.
//   Phase 2: float4 elementwise update of a (bandwidth-bound part).
__global__ __launch_bounds__(128)
void dadmm_sweep(const float* __restrict__ x, const float* __restrict__ y,
                 const float* __restrict__ hyp,
                 float* __restrict__ a, float* __restrict__ w,
                 const float* __restrict__ mu, const float* __restrict__ lam,
                 int k, int color) {
    const int tid  = threadIdx.x;
    const int lane = tid & 31;
    const int wave = tid >> 5;
    const int bg   = blockIdx.x;              // batch group 0..15
    const int p    = 2 * (int)blockIdx.y + color;
    const int pm   = (p + PN - 1) & (PN - 1);
    const int pp   = (p + 1) & (PN - 1);

    const float h0 = fabsf(hyp[k * 6 + 0]);
    const float h1 = fabsf(hyp[k * 6 + 1]);
    const float h2 = fabsf(hyp[k * 6 + 2]);
    const float h5 = fabsf(hyp[k * 6 + 5]);

    __shared__ float red[4][16];   // per-wave partial dots
    __shared__ float sbuf[16];     // s = xta + w - y per row

    // ----- phase 1: WMMA dot products -----
    // A operand (X, 16x4 f32): lane L -> row L%16, VGPR0/1 = K (2*half, 2*half+1)
    // B operand (a^T, 4x16 f32): identical per-lane addressing into `a`.
    const int row  = lane & 15;
    const int half = lane >> 4;
    const int b    = bg * 16 + row;
    const int base = (p * BSZ + b) * NF;
    const int k0   = wave * 196;               // this wave covers K in [k0, k0+196)

    // loop-invariant per-lane pointers; chunks advance by immediate offsets
    const float* xl = x + base + k0 + 2 * half;
    const float* al = a + base + k0 + 2 * half;

    v8f c = {0.f, 0.f, 0.f, 0.f, 0.f, 0.f, 0.f, 0.f};
#pragma unroll 1
    for (int kk = 0; kk < 7; ++kk) {
#pragma unroll
        for (int j = 0; j < 7; ++j) {          // offsets 0,16,...,96 bytes
            v2f av = *(const v2f*)(xl + j * 4);
            v2f bv = *(const v2f*)(al + j * 4);
            c = __builtin_amdgcn_wmma_f32_16x16x4_f32(false, av, false, bv,
                                                      (short)0, c, false, false);
        }
        xl += 28;
        al += 28;
    }
    // diagonal of C: row j<8 -> lane j, c[j]; row j>=8 -> lane j+16, c[j-8]
    {
        const int sel = lane & 7;
        float d = c[0];
#pragma unroll
        for (int i = 1; i < 8; ++i)
            if (sel == i) d = c[i];
        if (lane < 8)        red[wave][lane]      = d;
        else if (lane >= 24) red[wave][lane - 16] = d;
    }
    __syncthreads();

    // omega update (uses pre-sweep a via xta, pre-update w) + broadcast s
    if (tid < 16) {
        const int bb  = bg * 16 + tid;
        const int wi  = p * BSZ + bb;
        const float xta = red[0][tid] + red[1][tid] + red[2][tid] + red[3][tid];
        const float wp = w[wi];
        const float yp = y[wi];
        const float s  = xta + wp - yp;
        sbuf[tid] = s;
        const float wm = w[pm * BSZ + bb];
        const float wq = w[pp * BSZ + bb];
        const float lm = lam[wi];
        w[wi] = wp - h5 * (s + h2 * (2.f * wp - wm - wq) + 2.f * lm);
    }
    __syncthreads();

    // ----- phase 2: a update, float4 vector memory ops -----
    for (int r = 0; r < 16; ++r) {
        const float sr = sbuf[r];
        const int br = bg * 16 + r;
        const int b0 = (p  * BSZ + br) * NF;
        const int bm = (pm * BSZ + br) * NF;
        const int bq = (pp * BSZ + br) * NF;
        const float4* x4  = (const float4*)(x  + b0);
        float4*       a4  = (float4*)      (a  + b0);
        const float4* am4 = (const float4*)(a  + bm);
        const float4* aq4 = (const float4*)(a  + bq);
        const float4* m4  = (const float4*)(mu + b0);
        for (int n = tid; n < NF / 4; n += 128) {
            const float4 xv  = x4[n];
            const float4 av  = a4[n];
            const float4 amv = am4[n];
            const float4 aqv = aq4[n];
            const float4 mv  = m4[n];
            float4 o;
            o.x = av.x - h1 * (xv.x * sr + h0 * (2.f * av.x - amv.x - aqv.x) + 2.f * mv.x);
            o.y = av.y - h1 * (xv.y * sr + h0 * (2.f * av.y - amv.y - aqv.y) + 2.f * mv.y);
            o.z = av.z - h1 * (xv.z * sr + h0 * (2.f * av.z - amv.z - aqv.z) + 2.f * mv.z);
            o.w = av.w - h1 * (xv.w * sr + h0 * (2.f * av.w - amv.w - aqv.w) + 2.f * mv.w);
            a4[n] = o;
        }
    }
}

// ---------------- dual ascent for mu (post-sweep a) ----------------
// grid = (196, 64 nodes), block = 256; float4 per thread -> 196*256 = NODE_ELEMS/4
__global__ __launch_bounds__(256)
void dadmm_dual_a(const float* __restrict__ a, float* __restrict__ mu,
                  const float* __restrict__ hyp, int k) {
    const float h3 = fabsf(hyp[k * 6 + 3]);
    const int p  = blockIdx.y;
    const int pm = (p + PN - 1) & (PN - 1);
    const int pp = (p + 1) & (PN - 1);
    const int idx = blockIdx.x * 256 + threadIdx.x;   // float4 idx within node
    const float4* ap = (const float4*)(a + p  * NODE_ELEMS);
    const float4* am = (const float4*)(a + pm * NODE_ELEMS);
    const float4* aq = (const float4*)(a + pp * NODE_ELEMS);
    float4* m = (float4*)(mu + p * NODE_ELEMS);
    const float4 av  = ap[idx];
    const float4 amv = am[idx];
    const float4 aqv = aq[idx];
    float4 mv = m[idx];
    mv.x += h3 * (2.f * av.x - amv.x - aqv.x);
    mv.y += h3 * (2.f * av.y - amv.y - aqv.y);
    mv.z += h3 * (2.f * av.z - amv.z - aqv.z);
    mv.w += h3 * (2.f * av.w - amv.w - aqv.w);
    m[idx] = mv;
}

// ---------------- dual ascent for lam ----------------
__global__ __launch_bounds__(256)
void dadmm_dual_w(const float* __restrict__ w, float* __restrict__ lam,
                  const float* __restrict__ hyp, int k) {
    const float h4 = fabsf(hyp[k * 6 + 4]);
    const int idx = blockIdx.x * 256 + threadIdx.x;   // < W_ELEMS
    const int p = idx >> 8;           // / BSZ
    const int b = idx & 255;          // % BSZ
    const int pm = (p + PN - 1) & (PN - 1);
    const int pp = (p + 1) & (PN - 1);
    lam[idx] += h4 * (2.f * w[idx] - w[pm * BSZ + b] - w[pp * BSZ + b]);
}

// ---------------- launcher ----------------
extern "C" void kernel_launch(void* const* d_in, const int* in_sizes, int n_in,
                              void* d_out, int out_size, void* d_ws, size_t ws_size,
                              hipStream_t stream) {
    const float* x   = (const float*)d_in[0];   // inputs  [P,B,N,1]
    const float* y   = (const float*)d_in[1];   // labels  [P,B,1,1]
    const float* hyp = (const float*)d_in[2];   // [K,6]
    const float* a0  = (const float*)d_in[3];   // [P,B,N,1]
    const float* w0  = (const float*)d_in[4];   // [P,B,1,1]
    // d_in[5] = neighbors (ring; recomputed analytically)

    float* a   = (float*)d_out;                 // output a lives here
    float* w   = a + A_ELEMS;                   // output omega follows
    float* mu  = (float*)d_ws;                  // dual for a
    float* lam = mu + A_ELEMS;                  // dual for omega

    dadmm_init<<<dim3(A_ELEMS / 4 / 256), 256, 0, stream>>>(a0, w0, a, w, mu, lam);

    for (int k = 0; k < KIT; ++k) {
        dadmm_sweep<<<dim3(16, 32), 128, 0, stream>>>(x, y, hyp, a, w, mu, lam, k, 0);
        dadmm_sweep<<<dim3(16, 32), 128, 0, stream>>>(x, y, hyp, a, w, mu, lam, k, 1);
        dadmm_dual_a<<<dim3(NODE_ELEMS / 4 / 256, PN), 256, 0, stream>>>(a, mu, hyp, k);
        dadmm_dual_w<<<dim3(W_ELEMS / 256), 256, 0, stream>>>(w, lam, hyp, k);
    }
}